// SimpleVectorQuantizer_56899726737754
// MI455X (gfx1250) — compile-verified
//
#include <hip/hip_runtime.h>
#include <hip/hip_bf16.h>
#include <float.h>

// CDNA5 / gfx1250 vector-quantizer.
// dist(m,n) = ||z_m||^2 + ||e_n||^2 - 2 z_m.e_n ; argmin over n is independent
// of ||z_m||^2, so we minimize  en2[n] - 2*(z e^T)[m,n]  with the GEMM on the
// f32 WMMA pipe (v_wmma_f32_16x16x4_f32), exact f32 numerics.
// Blocking: 64 z-rows per block (4 x 16-row WMMA sub-tiles per wave) so each
// B (codebook) fragment read from L2 feeds 4 WMMAs -> 4x less L2 traffic than
// a 16-row block.

typedef float v2f __attribute__((ext_vector_type(2)));
typedef float v8f __attribute__((ext_vector_type(8)));

#define BATCH   32768
#define NE      8192
#define EDIM    256
#define TM      64            // z rows per block
#define MSUB    (TM / 16)     // 4 WMMA row sub-tiles per wave
#define TN      16            // codes per WMMA tile
#define NWAVES  8             // 256 threads / wave32
#define NTILES  (NE / TN)     // 512 codebook tiles
#define NBLK    (BATCH / TM)  // 512 blocks

// ---------------- kernel 1: codebook squared norms -> ws ----------------
__global__ __launch_bounds__(256) void vq_emb_norms(const float* __restrict__ emb,
                                                    float* __restrict__ norms) {
  __shared__ float red[256];
  const int row = blockIdx.x;
  const float v = emb[(size_t)row * EDIM + threadIdx.x];
  red[threadIdx.x] = v * v;
  __syncthreads();
  for (int s = 128; s > 0; s >>= 1) {
    if (threadIdx.x < s) red[threadIdx.x] += red[threadIdx.x + s];
    __syncthreads();
  }
  if (threadIdx.x == 0) norms[row] = red[0];
}

// ---------------- kernel 2: distances via WMMA + argmin + gather ----------------
__global__ __launch_bounds__(256)
void vq_main(const float* __restrict__ z, const float* __restrict__ emb,
             const float* __restrict__ enorm,
             float* __restrict__ zq_out, float* __restrict__ idx_out,
             float* __restrict__ partial) {
  __shared__ float zTile[TM][EDIM];        // 64 KB staged z rows
  __shared__ float candVal[TM][128];       // per-row candidates (8 waves x 16 lanes)
  __shared__ int   candIdx[TM][128];
  __shared__ int   rowBest[TM];
  __shared__ float red[256];

  const int tid   = threadIdx.x;
  const int wave  = tid >> 5;
  const int lane  = tid & 31;
  const int rbase = blockIdx.x * TM;

  // Stage the 64x256 z tile into LDS (float4 vectorized).
  {
    const float4* src = (const float4*)(z + (size_t)rbase * EDIM);
    float4* dst = (float4*)(&zTile[0][0]);
    for (int i = tid; i < TM * EDIM / 4; i += 256) dst[i] = src[i];
  }
  __syncthreads();

  const int l16   = lane & 15;          // M row within sub-tile (A) / N col (B,C)
  const int khalf = (lane >> 4) * 2;    // K sub-offset for 32-bit A/B fragments

  float minVal[MSUB][8];
  int   minIdx[MSUB][8];
#pragma unroll
  for (int s = 0; s < MSUB; ++s)
#pragma unroll
    for (int v = 0; v < 8; ++v) { minVal[s][v] = FLT_MAX; minIdx[s][v] = 0; }

  // Each wave sweeps codebook tiles t = wave, wave+8, ... (uniform trip count,
  // EXEC all ones inside -> legal for WMMA). Waves partition N; the block
  // collectively reads emb exactly once.
  for (int t = wave; t < NTILES; t += NWAVES) {
    const int nbase = t * TN;
    const int code  = nbase + l16;
    const float* bp = emb + (size_t)code * EDIM + khalf;

    if (t + NWAVES < NTILES)  // pull next tile toward L2/L0 (global_prefetch_b8)
      __builtin_prefetch(bp + (size_t)TN * NWAVES * EDIM, 0, 1);

    v8f acc[MSUB];
#pragma unroll
    for (int s = 0; s < MSUB; ++s) acc[s] = (v8f){0.f,0.f,0.f,0.f,0.f,0.f,0.f,0.f};

#pragma unroll 4
    for (int k0 = 0; k0 < EDIM; k0 += 4) {
      // B fragment: lane l holds emb[nbase + l%16][k0 + 2*(l/16) + {0,1}] (global b64)
      const float2 bv = *(const float2*)(bp + k0);
      v2f b; b[0] = bv.x; b[1] = bv.y;
#pragma unroll
      for (int s = 0; s < MSUB; ++s) {
        // A fragment: lane l holds z[rbase + s*16 + l%16][k0 + 2*(l/16) + {0,1}] (ds b64)
        const float2 av = *(const float2*)(&zTile[s * 16 + l16][k0 + khalf]);
        v2f a; a[0] = av.x; a[1] = av.y;
        // 8 args: (neg_a, A, neg_b, B, c_mod, C, reuse_a, reuse_b)
        acc[s] = __builtin_amdgcn_wmma_f32_16x16x4_f32(
            false, a, false, b, (short)0, acc[s], false, false);
      }
    }

    const float en = enorm[code];
#pragma unroll
    for (int s = 0; s < MSUB; ++s)
#pragma unroll
      for (int v = 0; v < 8; ++v) {
        // C layout: VGPR v, lanes 0-15 -> M=v, lanes 16-31 -> M=v+8; N = lane%16.
        const float d = en - 2.0f * acc[s][v];
        if (d < minVal[s][v]) { minVal[s][v] = d; minIdx[s][v] = code; }
      }
  }

  // Deposit per-lane candidates: row m = s*16 + v + 8*(lane>>4), slot = wave*16 + lane%16.
  {
    const int slot = wave * 16 + l16;
#pragma unroll
    for (int s = 0; s < MSUB; ++s)
#pragma unroll
      for (int v = 0; v < 8; ++v) {
        const int m = s * 16 + v + 8 * (lane >> 4);
        candVal[m][slot] = minVal[s][v];
        candIdx[m][slot] = minIdx[s][v];
      }
  }
  __syncthreads();

  // Final argmin per row (jnp.argmin tie-break: lowest index wins).
  if (tid < TM) {
    float bv = FLT_MAX;
    int   bi = NE;
    for (int s = 0; s < 128; ++s) {
      const float vv = candVal[tid][s];
      const int   ii = candIdx[tid][s];
      if (vv < bv || (vv == bv && ii < bi)) { bv = vv; bi = ii; }
    }
    rowBest[tid] = bi;
    idx_out[rbase + tid] = (float)bi;   // indices emitted as f32 values
  }
  __syncthreads();

  // Gather z_q = emb[best] (straight-through value) + per-block loss partial.
  float lsum = 0.f;
  for (int r = 0; r < TM; ++r) {
    const int   bi  = rowBest[r];
    const float c   = emb[(size_t)bi * EDIM + tid];
    const float zz  = zTile[r][tid];
    zq_out[(size_t)(rbase + r) * EDIM + tid] = c;
    const float dif = c - zz;
    lsum += dif * dif;
  }
  red[tid] = lsum;
  __syncthreads();
  for (int s = 128; s > 0; s >>= 1) {
    if (tid < s) red[tid] += red[tid + s];
    __syncthreads();
  }
  if (tid == 0) partial[blockIdx.x] = red[0];
}

// ---------------- kernel 3: deterministic loss reduction ----------------
__global__ __launch_bounds__(256) void vq_finalize(const float* __restrict__ partial,
                                                   float* __restrict__ lossOut) {
  __shared__ float red[256];
  float s = 0.f;
  for (int i = threadIdx.x; i < NBLK; i += 256) s += partial[i];
  red[threadIdx.x] = s;
  __syncthreads();
  for (int st = 128; st > 0; st >>= 1) {
    if (threadIdx.x < st) red[threadIdx.x] += red[threadIdx.x + st];
    __syncthreads();
  }
  if (threadIdx.x == 0)
    lossOut[0] = 1.25f * red[0] / ((float)BATCH * (float)EDIM);
}

extern "C" void kernel_launch(void* const* d_in, const int* in_sizes, int n_in,
                              void* d_out, int out_size, void* d_ws, size_t ws_size,
                              hipStream_t stream) {
  (void)in_sizes; (void)n_in; (void)out_size; (void)ws_size;
  const float* z   = (const float*)d_in[0];   // [32768, 256]
  const float* emb = (const float*)d_in[1];   // [8192, 256]

  float* out   = (float*)d_out;
  float* zq    = out;                                   // 32768*256
  float* loss  = out + (size_t)BATCH * EDIM;            // 1
  float* idxs  = loss + 1;                              // 32768

  float* norms   = (float*)d_ws;                        // 8192 floats
  float* partial = norms + NE;                          // 512 floats

  vq_emb_norms<<<NE, 256, 0, stream>>>(emb, norms);
  vq_main<<<NBLK, 256, 0, stream>>>(z, emb, norms, zq, idxs, partial);
  vq_finalize<<<1, 256, 0, stream>>>(partial, loss);
}